// Qwen3MoeLikeBlock_14894946583062
// MI455X (gfx1250) — compile-verified
//
#include <hip/hip_runtime.h>

typedef float v2f __attribute__((ext_vector_type(2)));
typedef float v8f __attribute__((ext_vector_type(8)));

#define T_TOKENS 8192
#define H_DIM    2048
#define E_EXP    8

#define BM  64
#define BN  128
#define BK  32
#define LDK 36   // padded LDS row stride (floats): 144B rows -> 16B aligned, conflict-free frag reads
#define KT_TILES (H_DIM / BK)          // 64
#define TOT_TILES (E_EXP * KT_TILES)   // 512

// ---------------- Router: logits, softmax/top-2, dense combine weights ----------------
__global__ __launch_bounds__(256) void moe_router_kernel(
    const float* __restrict__ x, const float* __restrict__ gate_w,
    float* __restrict__ logits_out, float* __restrict__ cw)
{
  const int wave  = threadIdx.x >> 5;
  const int lane  = threadIdx.x & 31;
  const int token = blockIdx.x * 8 + wave;
  if (token >= T_TOKENS) return;

  const float* xr = x + (size_t)token * H_DIM;
  float acc[E_EXP];
#pragma unroll
  for (int e = 0; e < E_EXP; ++e) acc[e] = 0.f;

  for (int h = lane; h < H_DIM; h += 32) {
    const float xv = xr[h];
#pragma unroll
    for (int e = 0; e < E_EXP; ++e) acc[e] += xv * gate_w[e * H_DIM + h];
  }
#pragma unroll
  for (int e = 0; e < E_EXP; ++e) {
#pragma unroll
    for (int off = 16; off > 0; off >>= 1)
      acc[e] += __shfl_xor(acc[e], off, 32);
  }
  if (lane == 0) {
#pragma unroll
    for (int e = 0; e < E_EXP; ++e) logits_out[token * E_EXP + e] = acc[e];
    int i1 = 0;
#pragma unroll
    for (int e = 1; e < E_EXP; ++e) if (acc[e] > acc[i1]) i1 = e;
    int i2 = (i1 == 0) ? 1 : 0;
#pragma unroll
    for (int e = 0; e < E_EXP; ++e) if (e != i1 && acc[e] > acc[i2]) i2 = e;
    // normalized top-2 softmax weights: softmax denom cancels under renorm
    const float w1 = 1.f / (1.f + __expf(acc[i2] - acc[i1]));
    const float w2 = 1.f - w1;
#pragma unroll
    for (int e = 0; e < E_EXP; ++e)
      cw[token * E_EXP + e] = (e == i1) ? w1 : ((e == i2) ? w2 : 0.f);
  }
}

// ---------------- MoE: out = sum_e cw[:,e] * (x @ W_e^T), fp32 WMMA ----------------
// Double-buffered tiles fed by async global->LDS DMA (ASYNCcnt), bypassing VGPRs.
__global__ __launch_bounds__(256) void moe_expert_kernel(
    const float* __restrict__ x, const float* __restrict__ expert_w,
    const float* __restrict__ cw, float* __restrict__ out)
{
  __shared__ float As[2][BM][LDK];     // x tile, [m][k]
  __shared__ float Bs[2][BN][LDK];     // W_e tile, [n][k]  (B = W_e^T -> B[k][n] = Bs[n][k])
  __shared__ float Cw[BM][E_EXP];      // combine weights for this token tile

  const int tid  = threadIdx.x;
  const int lane = tid & 31;
  const int wave = tid >> 5;
  const int wy   = wave >> 2;          // 0..1 : 32-row band
  const int wx   = wave & 3;           // 0..3 : 32-col band
  const int m0   = blockIdx.y * BM;
  const int n0   = blockIdx.x * BN;

  for (int idx = tid; idx < BM * E_EXP; idx += 256)
    ((float*)Cw)[idx] = cw[(size_t)m0 * E_EXP + idx];

  const int lrow = tid >> 3;           // 0..31
  const int lcol = (tid & 7) * 4;      // 0,4,...,28

  // Issue one tile's worth of async global->LDS loads (6 b128 ops per wave:
  // 2 for A, 4 for B). LDS dest addr = low 32 bits of generic pointer.
  auto issue_tile = [&](int t, int b) {
    const int e  = t >> 6;             // t / KT_TILES
    const int kt = (t & 63) * BK;
    const float* We = expert_w + (size_t)e * H_DIM * H_DIM;
#pragma unroll
    for (int p = 0; p < 2; ++p) {
      const int r = lrow + p * 32;
      const float* g = x + (size_t)(m0 + r) * H_DIM + kt + lcol;
      const unsigned l = (unsigned)(size_t)&As[b][r][lcol];
      asm volatile("global_load_async_to_lds_b128 %0, %1, off"
                   :: "v"(l), "v"(g) : "memory");
    }
#pragma unroll
    for (int p = 0; p < 4; ++p) {
      const int r = lrow + p * 32;
      const float* g = We + (size_t)(n0 + r) * H_DIM + kt + lcol;
      const unsigned l = (unsigned)(size_t)&Bs[b][r][lcol];
      asm volatile("global_load_async_to_lds_b128 %0, %1, off"
                   :: "v"(l), "v"(g) : "memory");
    }
  };

  v8f outacc[2][2] = {};
  v8f acc[2][2] = {};

  issue_tile(0, 0);                    // prologue
  int buf = 0;

  for (int t = 0; t < TOT_TILES; ++t) {
    __syncthreads();                   // everyone done reading buf^1 (tile t-1)
    if (t + 1 < TOT_TILES) {
      issue_tile(t + 1, buf ^ 1);
      // async loads complete in order: <=6 outstanding => tile t's batch landed
      asm volatile("s_wait_asynccnt 0x6" ::: "memory");
    } else {
      asm volatile("s_wait_asynccnt 0x0" ::: "memory");
    }
    __syncthreads();                   // all waves' tile-t data visible

#pragma unroll
    for (int kk = 0; kk < BK; kk += 4) {
      const int kb = kk + ((lane >> 4) << 1);   // lanes 0-15: K {0,1}; 16-31: K {2,3}
      v2f afrag[2], bfrag[2];
#pragma unroll
      for (int mi = 0; mi < 2; ++mi) {
        const int m = (lane & 15) + mi * 16 + wy * 32;
        afrag[mi].x = As[buf][m][kb];
        afrag[mi].y = As[buf][m][kb + 1];
      }
#pragma unroll
      for (int ni = 0; ni < 2; ++ni) {
        const int n = (lane & 15) + ni * 16 + wx * 32;
        bfrag[ni].x = Bs[buf][n][kb];
        bfrag[ni].y = Bs[buf][n][kb + 1];
      }
#pragma unroll
      for (int mi = 0; mi < 2; ++mi)
#pragma unroll
        for (int ni = 0; ni < 2; ++ni)
          acc[mi][ni] = __builtin_amdgcn_wmma_f32_16x16x4_f32(
              false, afrag[mi], false, bfrag[ni],
              (short)0, acc[mi][ni], false, false);
    }
    buf ^= 1;

    if ((t & 63) == 63) {
      // expert boundary: fold expert output into final accumulator, scaled by cw
      const int e = t >> 6;
#pragma unroll
      for (int mi = 0; mi < 2; ++mi) {
        const int mbase = ((lane >> 4) << 3) + mi * 16 + wy * 32;
#pragma unroll
        for (int i = 0; i < 8; ++i) {
          const float w = Cw[mbase + i][e];
#pragma unroll
          for (int ni = 0; ni < 2; ++ni)
            outacc[mi][ni][i] += w * acc[mi][ni][i];
        }
      }
#pragma unroll
      for (int mi = 0; mi < 2; ++mi)
#pragma unroll
        for (int ni = 0; ni < 2; ++ni)
          acc[mi][ni] = (v8f){};
    }
  }

  // single final store — no output pre-init required
#pragma unroll
  for (int mi = 0; mi < 2; ++mi) {
    const int mbase = m0 + ((lane >> 4) << 3) + mi * 16 + wy * 32;
#pragma unroll
    for (int ni = 0; ni < 2; ++ni) {
      const int n = n0 + (lane & 15) + ni * 16 + wx * 32;
#pragma unroll
      for (int i = 0; i < 8; ++i)
        out[(size_t)(mbase + i) * H_DIM + n] = outacc[mi][ni][i];
    }
  }
}

extern "C" void kernel_launch(void* const* d_in, const int* in_sizes, int n_in,
                              void* d_out, int out_size, void* d_ws, size_t ws_size,
                              hipStream_t stream) {
  (void)in_sizes; (void)n_in; (void)out_size; (void)ws_size;
  const float* x        = (const float*)d_in[0];   // [T, H]
  const float* gate_w   = (const float*)d_in[1];   // [E, H]
  const float* expert_w = (const float*)d_in[2];   // [E, H, H]
  float* out    = (float*)d_out;                               // [T, H]
  float* logits = out + (size_t)T_TOKENS * H_DIM;              // [T, E] tail
  float* cw     = (float*)d_ws;                                // [T, E] scratch (256 KB)

  moe_router_kernel<<<T_TOKENS / 8, 256, 0, stream>>>(x, gate_w, logits, cw);

  dim3 grid(H_DIM / BN, T_TOKENS / BM);
  moe_expert_kernel<<<grid, 256, 0, stream>>>(x, expert_w, cw, out);
}